// NTM_40776419508905
// MI455X (gfx1250) — compile-verified
//
#include <hip/hip_runtime.h>
#include <math.h>

#define B 64
#define T 64
#define IN_DIM 256
#define H 512
#define NN 4096
#define MM 128
#define EPSF 1e-8f
#define NPB 256            // memory rows per sweep block
#define CHUNKS (NN / NPB)  // 16

typedef __attribute__((ext_vector_type(2))) float v2f;
typedef __attribute__((ext_vector_type(4))) float f4;
typedef __attribute__((ext_vector_type(8))) float v8f;

// ---------------- wave / block reductions (fixed tree -> deterministic) ---------

__device__ inline float waveReduceSum(float v) {
    #pragma unroll
    for (int off = 16; off > 0; off >>= 1) v += __shfl_xor(v, off, 32);
    return v;
}

__device__ inline float waveReduceMax(float v) {
    #pragma unroll
    for (int off = 16; off > 0; off >>= 1) v = fmaxf(v, __shfl_xor(v, off, 32));
    return v;
}

__device__ inline float blockReduce(float v, float* red, bool isMax) {
    int lane = threadIdx.x & 31, wid = threadIdx.x >> 5;
    int nw = blockDim.x >> 5;
    v = isMax ? waveReduceMax(v) : waveReduceSum(v);
    if (lane == 0) red[wid] = v;
    __syncthreads();
    if (wid == 0) {
        float x = (lane < nw) ? red[lane] : (isMax ? -3.402823e38f : 0.f);
        x = isMax ? waveReduceMax(x) : waveReduceSum(x);
        if (lane == 0) red[0] = x;
    }
    __syncthreads();
    float r = red[0];
    __syncthreads();   // allow scratch reuse
    return r;
}

__device__ inline float softplusf(float x) {
    return (x > 20.f) ? x : log1pf(expf(x));
}
__device__ inline float sigmoidf(float x) {
    return 1.f / (1.f + expf(-x));
}

// ---------------- fp32 WMMA GEMM (one wave = one 16x16 tile, K-loop of 16x16x4) --

__device__ inline v8f wmma_span(const float* __restrict__ A, int lda,
                                const float* __restrict__ Wc, int ldw,
                                int K, v8f acc) {
    int lane = threadIdx.x & 31;
    int half = lane >> 4;     // 0: K=0,1 ; 1: K=2,3 (per CDNA5 32-bit A layout)
    int rr   = lane & 15;     // A row / B col within tile
    for (int k0 = 0; k0 < K; k0 += 4) {
        v2f av, bv;
        av.x = A[rr * lda + k0 + 2 * half + 0];
        av.y = A[rr * lda + k0 + 2 * half + 1];
        bv.x = Wc[(k0 + 2 * half + 0) * ldw + rr];
        bv.y = Wc[(k0 + 2 * half + 1) * ldw + rr];
        acc = __builtin_amdgcn_wmma_f32_16x16x4_f32(false, av, false, bv,
                                                    (short)0, acc, false, false);
    }
    return acc;
}

__device__ inline void wmma_store_act(v8f acc, const float* bias, float* out,
                                      int rowBase, int colBase, int Ncols, int act) {
    int lane = threadIdx.x & 31;
    int half = lane >> 4, cc = lane & 15;
    float bvv = bias ? bias[colBase + cc] : 0.f;
    #pragma unroll
    for (int v = 0; v < 8; ++v) {
        float x = acc[v] + bvv;
        if (act == 1)      x = tanhf(x);
        else if (act == 2) x = sigmoidf(x);
        out[(size_t)(rowBase + v + 8 * half) * Ncols + colBase + cc] = x;
    }
}

// controller-style: out = act(A@W + A2@W2 + bias). grid=(4, Ncols/16), block=32.
__global__ void gemm_act_kernel(const float* __restrict__ A, int lda,
                                const float* __restrict__ W, int K,
                                const float* __restrict__ A2, int lda2,
                                const float* __restrict__ W2, int K2,
                                const float* __restrict__ bias,
                                float* __restrict__ out, int Ncols, int act) {
    int rowBase = blockIdx.x * 16;
    int colBase = blockIdx.y * 16;
    v8f acc = {};
    acc = wmma_span(A + (size_t)rowBase * lda, lda, W + colBase, Ncols, K, acc);
    if (A2 != nullptr)
        acc = wmma_span(A2 + (size_t)rowBase * lda2, lda2, W2 + colBase, Ncols, K2, acc);
    wmma_store_act(acc, bias, out, rowBase, colBase, Ncols, act);
}

// dual-output: blockIdx.z selects {W,bias,out,act}. grid=(4, Ncols/16, 2), block=32.
__global__ void gemm_act_dual_kernel(const float* __restrict__ A, int lda,
                                     const float* __restrict__ W0,
                                     const float* __restrict__ W1, int K,
                                     const float* __restrict__ bias0,
                                     const float* __restrict__ bias1,
                                     float* __restrict__ out0,
                                     float* __restrict__ out1,
                                     int Ncols, int act0, int act1) {
    int rowBase = blockIdx.x * 16;
    int colBase = blockIdx.y * 16;
    const float* W    = blockIdx.z ? W1 : W0;      // uniform branch, EXEC stays full
    const float* bias = blockIdx.z ? bias1 : bias0;
    float* out        = blockIdx.z ? out1 : out0;
    int act           = blockIdx.z ? act1 : act0;
    v8f acc = {};
    acc = wmma_span(A + (size_t)rowBase * lda, lda, W + colBase, Ncols, K, acc);
    wmma_store_act(acc, bias, out, rowBase, colBase, Ncols, act);
}

// ---------------- fused memory sweep: erase/add + r partials + row norms --------
// grid=(B, CHUNKS), block=256 (8 waves); lane owns 4 consecutive m (float4).
__global__ void sweep_kernel(float* __restrict__ mem,
                             const float* __restrict__ wr,
                             const float* __restrict__ ww,
                             const float* __restrict__ e,
                             const float* __restrict__ a,
                             float* __restrict__ r_part,
                             float* __restrict__ mnorm) {
    __shared__ float lds[8 * MM];
    int b = blockIdx.x;
    int nBase = blockIdx.y * NPB;
    int wave = threadIdx.x >> 5, lane = threadIdx.x & 31;
    f4 e4 = *(const f4*)(e + (size_t)b * MM + lane * 4);
    f4 a4 = *(const f4*)(a + (size_t)b * MM + lane * 4);
    f4 racc = {0.f, 0.f, 0.f, 0.f};
    #pragma unroll 2
    for (int i = 0; i < NPB / 8; ++i) {
        int n = nBase + wave * (NPB / 8) + i;
        float wwv = ww[(size_t)b * NN + n];
        float wrv = wr[(size_t)b * NN + n];
        float* mp = mem + ((size_t)b * NN + n) * MM + lane * 4;
        f4 m4 = *(const f4*)mp;
        m4 = m4 * (1.f - wwv * e4) + wwv * a4;
        *(f4*)mp = m4;
        racc += wrv * m4;
        float sq = m4.x * m4.x + m4.y * m4.y + m4.z * m4.z + m4.w * m4.w;
        sq = waveReduceSum(sq);
        if (lane == 0) mnorm[(size_t)b * NN + n] = sqrtf(sq);
    }
    lds[wave * MM + lane * 4 + 0] = racc.x;
    lds[wave * MM + lane * 4 + 1] = racc.y;
    lds[wave * MM + lane * 4 + 2] = racc.z;
    lds[wave * MM + lane * 4 + 3] = racc.w;
    __syncthreads();
    if (threadIdx.x < MM) {
        float s = 0.f;
        #pragma unroll
        for (int w = 0; w < 8; ++w) s += lds[w * MM + threadIdx.x];
        r_part[((size_t)b * CHUNKS + blockIdx.y) * MM + threadIdx.x] = s;
    }
}

__global__ void reduce_r_kernel(const float* __restrict__ r_part,
                                float* __restrict__ r) {
    int idx = blockIdx.x * blockDim.x + threadIdx.x;
    if (idx >= B * MM) return;
    int b = idx / MM, m = idx % MM;
    float s = 0.f;
    #pragma unroll
    for (int c = 0; c < CHUNKS; ++c) s += r_part[((size_t)b * CHUNKS + c) * MM + m];
    r[idx] = s;
}

// ---------------- fused addressing: dots + head params + softmax/shift/sharpen --
// grid=B, block=1024 (32 waves).
//   phase 1: content dots for both heads -> LDS (wave per memory row, coalesced)
//   phase 2: 12 tiny H-projections (one wave each) + 2 k-norms -> LDS
//   phase 3: per head: cosine softmax, gate, circular 3-tap shift, sharpen, norm
__global__ void addressing_kernel(const float* __restrict__ mem,
                                  const float* __restrict__ h,
                                  const float* __restrict__ k_r,
                                  const float* __restrict__ k_w,
                                  const float* __restrict__ r_Wbeta,
                                  const float* __restrict__ r_Wg,
                                  const float* __restrict__ r_Ws,
                                  const float* __restrict__ r_Wgam,
                                  const float* __restrict__ w_Wbeta,
                                  const float* __restrict__ w_Wg,
                                  const float* __restrict__ w_Ws,
                                  const float* __restrict__ w_Wgam,
                                  const float* __restrict__ mnorm,
                                  float* __restrict__ wvec_r,
                                  float* __restrict__ wvec_w) {
    __shared__ float dotA[NN];   // 16KB: dot with k_r
    __shared__ float dotB[NN];   // 16KB: dot with k_w
    __shared__ float wg[NN];     // 16KB: gated weights (reused per head)
    __shared__ float hp[16];     // raw head-param dots + k norms
    __shared__ float red[32];

    int b = blockIdx.x;
    int tid = threadIdx.x;
    int wave = tid >> 5, lane = tid & 31;

    f4 kr4 = *(const f4*)(k_r + (size_t)b * MM + lane * 4);
    f4 kw4 = *(const f4*)(k_w + (size_t)b * MM + lane * 4);

    // phase 1: dots (each wave owns 128 consecutive rows; 512B coalesced reads)
    {
        const float* base = mem + ((size_t)b * NN) * MM + lane * 4;
        #pragma unroll 4
        for (int i = 0; i < NN / 32; ++i) {
            int n = wave * (NN / 32) + i;
            const float* rowp = base + (size_t)n * MM;
            __builtin_prefetch(rowp + 4 * MM, 0, 3);   // pull row n+4 toward WGP
            f4 m4 = *(const f4*)rowp;
            float sr = m4.x * kr4.x + m4.y * kr4.y + m4.z * kr4.z + m4.w * kr4.w;
            float sw = m4.x * kw4.x + m4.y * kw4.y + m4.z * kw4.z + m4.w * kw4.w;
            sr = waveReduceSum(sr);
            sw = waveReduceSum(sw);
            if (lane == 0) { dotA[n] = sr; dotB[n] = sw; }
        }
    }

    // phase 2: head-parameter projections (waves 0..11) and k norms (waves 12,13)
    if (wave < 12) {
        int head = wave / 6;
        int j = wave - head * 6;
        const float* W;
        int stride = 1, cofs = 0;
        if (j == 0)      W = head ? w_Wbeta : r_Wbeta;
        else if (j == 1) W = head ? w_Wg   : r_Wg;
        else if (j <= 4) { W = head ? w_Ws : r_Ws; stride = 3; cofs = j - 2; }
        else             W = head ? w_Wgam : r_Wgam;
        const float* hb = h + (size_t)b * H;
        float s = 0.f;
        for (int i = lane; i < H; i += 32) s += hb[i] * W[i * stride + cofs];
        s = waveReduceSum(s);
        if (lane == 0) hp[wave] = s;
    } else if (wave == 12) {
        float s = kr4.x * kr4.x + kr4.y * kr4.y + kr4.z * kr4.z + kr4.w * kr4.w;
        s = waveReduceSum(s);
        if (lane == 0) hp[12] = sqrtf(s);
    } else if (wave == 13) {
        float s = kw4.x * kw4.x + kw4.y * kw4.y + kw4.z * kw4.z + kw4.w * kw4.w;
        s = waveReduceSum(s);
        if (lane == 0) hp[13] = sqrtf(s);
    }
    __syncthreads();

    // phase 3: both heads, sequential (uniform control flow, block reductions)
    for (int head = 0; head < 2; ++head) {
        float* wvec = head ? wvec_w : wvec_r;
        const float* dotp = head ? dotB : dotA;
        float beta  = softplusf(hp[head * 6 + 0]);
        float g     = sigmoidf(hp[head * 6 + 1]);
        float a0 = hp[head * 6 + 2], a1 = hp[head * 6 + 3], a2 = hp[head * 6 + 4];
        float mx3 = fmaxf(a0, fmaxf(a1, a2));
        float e0 = expf(a0 - mx3), e1 = expf(a1 - mx3), e2 = expf(a2 - mx3);
        float ssum = e0 + e1 + e2;
        float s0 = e0 / ssum, s1 = e1 / ssum, s2 = e2 / ssum;
        float gamma = 1.f + softplusf(hp[head * 6 + 5]);
        float knorm = hp[12 + head];

        float sc[4], wprev[4];
        float lmax = -3.402823e38f;
        #pragma unroll
        for (int j = 0; j < 4; ++j) {
            int n = tid + j * 1024;
            sc[j] = beta * dotp[n] / (mnorm[(size_t)b * NN + n] * knorm + EPSF);
            wprev[j] = wvec[(size_t)b * NN + n];
            lmax = fmaxf(lmax, sc[j]);
        }
        float Mx = blockReduce(lmax, red, true);
        float lsum = 0.f;
        #pragma unroll
        for (int j = 0; j < 4; ++j) { sc[j] = expf(sc[j] - Mx); lsum += sc[j]; }
        float S = blockReduce(lsum, red, false);
        float invS = 1.f / S;
        #pragma unroll
        for (int j = 0; j < 4; ++j) {
            int n = tid + j * 1024;
            wg[n] = g * sc[j] * invS + (1.f - g) * wprev[j];
        }
        __syncthreads();
        float wp[4];
        float lsum2 = 0.f;
        #pragma unroll
        for (int j = 0; j < 4; ++j) {
            int n = tid + j * 1024;
            float wt = s0 * wg[(n + 1) & (NN - 1)] + s1 * wg[n] +
                       s2 * wg[(n - 1 + NN) & (NN - 1)];
            wp[j] = powf(wt + EPSF, gamma);
            lsum2 += wp[j];
        }
        float S2 = blockReduce(lsum2, red, false);
        float inv2 = 1.f / S2;
        #pragma unroll
        for (int j = 0; j < 4; ++j)
            wvec[(size_t)b * NN + tid + j * 1024] = wp[j] * inv2;
        __syncthreads();   // wg reused by next head
    }
}

__global__ void init_kernel(float* __restrict__ wr, float* __restrict__ ww,
                            float* __restrict__ h) {
    int idx = blockIdx.x * blockDim.x + threadIdx.x;
    if (idx < B * NN) {
        float v = (idx == 0) ? 1.f : 0.f;   // w0[0,0] = 1, all else 0
        wr[idx] = v;
        ww[idx] = v;
    }
    if (idx < B * H) h[idx] = 0.f;
}

// -------------------------------- launch ---------------------------------------

extern "C" void kernel_launch(void* const* d_in, const int* in_sizes, int n_in,
                              void* d_out, int out_size, void* d_ws, size_t ws_size,
                              hipStream_t stream) {
    const float* x      = (const float*)d_in[0];
    const float* mem0   = (const float*)d_in[1];
    const float* Wx     = (const float*)d_in[2];
    const float* Wr     = (const float*)d_in[3];
    const float* bC     = (const float*)d_in[4];
    const float* r_Wk   = (const float*)d_in[5];
    const float* r_bk   = (const float*)d_in[6];
    const float* r_Wbeta= (const float*)d_in[7];
    const float* r_Wg   = (const float*)d_in[8];
    const float* r_Ws   = (const float*)d_in[9];
    const float* r_Wgam = (const float*)d_in[10];
    const float* w_Wk   = (const float*)d_in[11];
    const float* w_bk   = (const float*)d_in[12];
    const float* w_Wbeta= (const float*)d_in[13];
    const float* w_Wg   = (const float*)d_in[14];
    const float* w_Ws   = (const float*)d_in[15];
    const float* w_Wgam = (const float*)d_in[16];
    const float* w_We   = (const float*)d_in[17];
    const float* w_Wa   = (const float*)d_in[18];

    float* ws = (float*)d_ws;
    size_t off = 0;
    float* mem    = ws + off; off += (size_t)B * NN * MM;   // 128 MB, L2-resident
    float* wr     = ws + off; off += (size_t)B * NN;
    float* wwv    = ws + off; off += (size_t)B * NN;
    float* h      = ws + off; off += (size_t)B * H;
    float* e      = ws + off; off += (size_t)B * MM;
    float* a      = ws + off; off += (size_t)B * MM;
    float* kr     = ws + off; off += (size_t)B * MM;
    float* kw     = ws + off; off += (size_t)B * MM;
    float* r      = ws + off; off += (size_t)B * MM;
    float* r_part = ws + off; off += (size_t)B * CHUNKS * MM;
    float* mnorm  = ws + off; off += (size_t)B * NN;

    hipMemcpyAsync(mem, mem0, (size_t)B * NN * MM * sizeof(float),
                   hipMemcpyDeviceToDevice, stream);
    init_kernel<<<(B * NN + 255) / 256, 256, 0, stream>>>(wr, wwv, h);

    for (int t = 0; t < T; ++t) {
        // e = sigmoid(h @ w_We), a = tanh(h @ w_Wa)   (uses previous h)
        gemm_act_dual_kernel<<<dim3(4, MM / 16, 2), 32, 0, stream>>>(
            h, H, w_We, w_Wa, H, nullptr, nullptr, e, a, MM, 2, 1);
        // fused: mem erase/add, r partials, row norms of new mem
        sweep_kernel<<<dim3(B, CHUNKS), 256, 0, stream>>>(
            mem, wr, wwv, e, a, r_part, mnorm);
        reduce_r_kernel<<<(B * MM + 255) / 256, 256, 0, stream>>>(r_part, r);
        // h = tanh(x_t @ Wx + r @ Wr + bC)
        gemm_act_kernel<<<dim3(4, H / 16), 32, 0, stream>>>(
            x + (size_t)t * IN_DIM, T * IN_DIM, Wx, IN_DIM,
            r, MM, Wr, MM, bC, h, H, 1);
        // k_r = tanh(h @ r_Wk + r_bk), k_w = tanh(h @ w_Wk + w_bk)
        gemm_act_dual_kernel<<<dim3(4, MM / 16, 2), 32, 0, stream>>>(
            h, H, r_Wk, w_Wk, H, r_bk, w_bk, kr, kw, MM, 1, 1);
        // fused dots + head params + both addressing updates
        addressing_kernel<<<B, 1024, 0, stream>>>(
            mem, h, kr, kw,
            r_Wbeta, r_Wg, r_Ws, r_Wgam,
            w_Wbeta, w_Wg, w_Ws, w_Wgam,
            mnorm, wr, wwv);
    }

    hipMemcpyAsync(d_out, h, (size_t)B * H * sizeof(float),
                   hipMemcpyDeviceToDevice, stream);
}